// GCN_41626823032948
// MI455X (gfx1250) — compile-verified
//
#include <hip/hip_runtime.h>
#include <hip/hip_bf16.h>
#include <stdint.h>

// ---------------------------------------------------------------------------
// GCN 2-layer forward for MI455X (gfx1250).
//   Layer: h = X @ W (fp32 WMMA 16x16x4);  out = scatter(norm * h[row] -> col)
//          + self-loop + bias;  ReLU between layers.
// Dense GEMMs use V_WMMA_F32_16X16X4_F32 (exact fp32). Aggregation uses
// native f32 global atomics (L2-resident working set: 192MB L2 >> 128MB).
// ---------------------------------------------------------------------------

typedef float v2f __attribute__((ext_vector_type(2)));
typedef float v4f __attribute__((ext_vector_type(4)));
typedef float v8f __attribute__((ext_vector_type(8)));

#define N_NODES 100000
#define N_EDGES 3200000
#define F_IN    256
#define F_H     128
#define F_OUT   64

using eidx_t = long long;   // reference edge_index dtype is int64

// ---------------------------- degree kernels -------------------------------

__global__ void deg_init_kernel(float* __restrict__ deg) {
    int i = blockIdx.x * blockDim.x + threadIdx.x;
    if (i < N_NODES) deg[i] = 1.0f;              // self-loop contributes 1
}

__global__ void deg_count_kernel(const eidx_t* __restrict__ col,
                                 float* __restrict__ deg) {
    int i = blockIdx.x * blockDim.x + threadIdx.x;
    if (i < N_EDGES) unsafeAtomicAdd(&deg[(int)col[i]], 1.0f);
}

__global__ void deg_rsqrt_kernel(float* __restrict__ deg) {
    int i = blockIdx.x * blockDim.x + threadIdx.x;
    if (i < N_NODES) {
        float d = deg[i];
        deg[i] = (d > 0.0f) ? rsqrtf(d) : 0.0f;  // dis = deg^{-1/2}
    }
}

// ------------------------------ fp32 WMMA GEMM -----------------------------
// One wave computes a 16x16 tile of Out = A[M x K] @ W[K x NCOL].
// blockIdx.x -> M-tile (16 rows), wave id -> 16-column slice.
// A frag (16x4 f32): lanes 0-15 hold M=0..15 K={0,1}; lanes 16-31 K={2,3}.
// B frag (4x16 f32): lanes 0-15 hold N=0..15 K={0,1}; lanes 16-31 K={2,3}.
// C/D (16x16 f32):   VGPR i -> M = i + 8*(lane>=16), N = lane%16.

template <int K, int NCOL>
__global__ __launch_bounds__((NCOL / 16) * 32)
void wmma_gemm_kernel(const float* __restrict__ A,
                      const float* __restrict__ W,
                      float* __restrict__ Out) {
    const int lane = threadIdx.x & 31;
    const int wave = threadIdx.x >> 5;
    const int m0   = blockIdx.x * 16;
    const int n0   = wave * 16;
    const int mr   = lane & 15;   // row (A) / col (B) within tile
    const int hi   = lane >> 4;   // K half-select: 0 -> K{0,1}, 1 -> K{2,3}

    const float* __restrict__ arow = A + (size_t)(m0 + mr) * K;
    const float* __restrict__ wcol = W + n0 + mr;

    v8f acc = {0.f, 0.f, 0.f, 0.f, 0.f, 0.f, 0.f, 0.f};

#pragma unroll 4
    for (int k = 0; k < K; k += 4) {
        const int ka = k + 2 * hi;
        v2f a = *(const v2f*)(arow + ka);           // contiguous, 8B aligned
        v2f b;
        b.x = wcol[(size_t)ka * NCOL];
        b.y = wcol[(size_t)(ka + 1) * NCOL];
        // (neg_a, A, neg_b, B, c_mod, C, reuse_a, reuse_b)
        acc = __builtin_amdgcn_wmma_f32_16x16x4_f32(
            false, a, false, b, (short)0, acc, false, false);
    }

    float* __restrict__ orow = Out + (size_t)(m0 + 8 * hi) * NCOL + n0 + mr;
#pragma unroll
    for (int i = 0; i < 8; ++i)
        orow[(size_t)i * NCOL] = acc[i];
}

// --------------------------- aggregation kernels ---------------------------
// out = bias + dis[i]^2 * h[i]   (self-loop term; dis^2 == 1/deg)

template <int F>
__global__ void agg_init_kernel(const float* __restrict__ h,
                                const float* __restrict__ bias,
                                const float* __restrict__ dis,
                                float* __restrict__ out) {
    size_t i = (size_t)blockIdx.x * blockDim.x + threadIdx.x;
    if (i >= (size_t)N_NODES * F) return;
    const int node = (int)(i / F);
    const int f    = (int)(i & (F - 1));
    const float d  = dis[node];
    out[i] = bias[f] + d * d * h[i];
}

// out[col] += dis[row]*dis[col] * h[row]   -- one wave per edge.
template <int F>
__global__ __launch_bounds__(256)
void agg_edge_kernel(const eidx_t* __restrict__ row,
                     const eidx_t* __restrict__ col,
                     const float* __restrict__ dis,
                     const float* __restrict__ h,
                     float* __restrict__ out) {
    constexpr int VPT = F / 32;                 // floats per lane (4 or 2)
    const int lane = threadIdx.x & 31;
    const int wave = threadIdx.x >> 5;
    const long long e = (long long)blockIdx.x * 8 + wave;
    if (e >= N_EDGES) return;

    const int r = (int)row[e];
    const int c = (int)col[e];
    const float w = dis[r] * dis[c];

    const float* __restrict__ hp = h + (size_t)r * F + lane * VPT;
    float* __restrict__ op = out + (size_t)c * F + lane * VPT;

    if constexpr (VPT == 4) {
        v4f hv = *(const v4f*)hp;               // global_load_b128
        unsafeAtomicAdd(op + 0, w * hv.x);
        unsafeAtomicAdd(op + 1, w * hv.y);
        unsafeAtomicAdd(op + 2, w * hv.z);
        unsafeAtomicAdd(op + 3, w * hv.w);
    } else {
        v2f hv = *(const v2f*)hp;               // global_load_b64
        unsafeAtomicAdd(op + 0, w * hv.x);
        unsafeAtomicAdd(op + 1, w * hv.y);
    }
}

__global__ void relu_kernel(float* __restrict__ a, size_t n) {
    size_t i = (size_t)blockIdx.x * blockDim.x + threadIdx.x;
    if (i < n) a[i] = fmaxf(a[i], 0.0f);
}

// ------------------------------- launcher ----------------------------------

extern "C" void kernel_launch(void* const* d_in, const int* in_sizes, int n_in,
                              void* d_out, int out_size, void* d_ws, size_t ws_size,
                              hipStream_t stream) {
    const float*  x   = (const float*)d_in[0];
    const eidx_t* ei  = (const eidx_t*)d_in[1];
    const float*  W1  = (const float*)d_in[2];
    const float*  b1  = (const float*)d_in[3];
    const float*  W2  = (const float*)d_in[4];
    const float*  b2  = (const float*)d_in[5];
    float* out = (float*)d_out;

    const eidx_t* row = ei;              // edge_index[0]
    const eidx_t* col = ei + N_EDGES;    // edge_index[1]

    // workspace: dis[N] | h[N*F_H] (reused for layer-2 h) | a1[N*F_H]
    float* dis = (float*)d_ws;
    float* h1  = dis + N_NODES;
    float* a1  = h1 + (size_t)N_NODES * F_H;

    // --- degree / normalization ---
    deg_init_kernel<<<(N_NODES + 255) / 256, 256, 0, stream>>>(dis);
    deg_count_kernel<<<(N_EDGES + 255) / 256, 256, 0, stream>>>(col, dis);
    deg_rsqrt_kernel<<<(N_NODES + 255) / 256, 256, 0, stream>>>(dis);

    // --- layer 1: h1 = x @ W1 ; a1 = aggregate + b1 ; ReLU ---
    wmma_gemm_kernel<F_IN, F_H>
        <<<N_NODES / 16, (F_H / 16) * 32, 0, stream>>>(x, W1, h1);
    {
        size_t tot = (size_t)N_NODES * F_H;
        agg_init_kernel<F_H>
            <<<(unsigned)((tot + 255) / 256), 256, 0, stream>>>(h1, b1, dis, a1);
    }
    agg_edge_kernel<F_H>
        <<<(N_EDGES + 7) / 8, 256, 0, stream>>>(row, col, dis, h1, a1);
    {
        size_t tot = (size_t)N_NODES * F_H;
        relu_kernel<<<(unsigned)((tot + 255) / 256), 256, 0, stream>>>(a1, tot);
    }

    // --- layer 2: h2 = a1 @ W2 (reuse h1 buffer) ; out = aggregate + b2 ---
    wmma_gemm_kernel<F_H, F_OUT>
        <<<N_NODES / 16, (F_OUT / 16) * 32, 0, stream>>>(a1, W2, h1);
    {
        size_t tot = (size_t)N_NODES * F_OUT;
        agg_init_kernel<F_OUT>
            <<<(unsigned)((tot + 255) / 256), 256, 0, stream>>>(h1, b2, dis, out);
    }
    agg_edge_kernel<F_OUT>
        <<<(N_EDGES + 7) / 8, 256, 0, stream>>>(row, col, dis, h1, out);
}